// Decoder_67190468378803
// MI455X (gfx1250) — compile-verified
//
#include <hip/hip_runtime.h>
#include <hip/hip_bf16.h>
#include <math.h>

typedef __attribute__((ext_vector_type(2))) float v2f;
typedef __attribute__((ext_vector_type(8))) float v8f;

#define HDIM 128

__device__ __forceinline__ v8f wmma_f32_k4(v2f a, v2f b, v8f c) {
  // D = A(16x4,f32) * B(4x16,f32) + C(16x16,f32)
  return __builtin_amdgcn_wmma_f32_16x16x4_f32(
      /*neg_a=*/false, a, /*neg_b=*/false, b,
      /*c_mod=*/(short)0, c, /*reuse_a=*/false, /*reuse_b=*/false);
}

__device__ __forceinline__ float sigmoidf_(float x) {
  return 1.0f / (1.0f + __expf(-x));
}

// ---------------------------------------------------------------------------
// LSTM gate GEMM: h[n,j] = sig(o)*tanh(sig(i)*tanh(g)),
// gates = z @ W_ih^T + (b_ih + b_hh).  The f-gate is dead (c0 == 0) -> skipped.
// One wave = 16 node rows; block = 8 waves = 128 rows. A fragment is reused by
// the three gate WMMAs per K-step (good clause/co-exec shape, verified in asm).
// ---------------------------------------------------------------------------
__global__ __launch_bounds__(256) void lstm_kernel(
    const float* __restrict__ z, const float* __restrict__ Wih,
    const float* __restrict__ bih, const float* __restrict__ bhh,
    float* __restrict__ h, int n) {
  const int lane = threadIdx.x & 31;
  const int wave = threadIdx.x >> 5;
  const int rowbase = blockIdx.x * 128 + wave * 16;
  if (rowbase >= n) return;                       // wave-uniform exit, no barriers
  const int mrow  = lane & 15;                    // A-row / B-col / D-col lane id
  const int khalf = (lane >> 4) << 1;             // 0 or 2
  int arow = rowbase + mrow;
  if (arow >= n) arow = n - 1;                    // clamp loads; stores guarded
  const float* aptr = z + (size_t)arow * HDIM + khalf;

  for (int c = 0; c < 8; ++c) {                   // 8 column chunks of 16
    const int jcol = c * 16 + mrow;
    const float* bi_ptr = Wih + ((size_t)(jcol      )) * HDIM + khalf; // i gate
    const float* bg_ptr = Wih + ((size_t)(256 + jcol)) * HDIM + khalf; // g gate
    const float* bo_ptr = Wih + ((size_t)(384 + jcol)) * HDIM + khalf; // o gate

    v8f acc_i = {}, acc_g = {}, acc_o = {};
    for (int k = 0; k < HDIM; k += 4) {
      v2f a  = *(const v2f*)(aptr + k);
      v2f bi = *(const v2f*)(bi_ptr + k);
      v2f bg = *(const v2f*)(bg_ptr + k);
      v2f bo = *(const v2f*)(bo_ptr + k);
      acc_i = wmma_f32_k4(a, bi, acc_i);
      acc_g = wmma_f32_k4(a, bg, acc_g);
      acc_o = wmma_f32_k4(a, bo, acc_o);
    }
    const float Bi = bih[jcol]       + bhh[jcol];
    const float Bg = bih[256 + jcol] + bhh[256 + jcol];
    const float Bo = bih[384 + jcol] + bhh[384 + jcol];
    const int mofs = (lane >> 4) << 3;            // 0 or 8
    #pragma unroll
    for (int r = 0; r < 8; ++r) {
      const int row = rowbase + r + mofs;
      if (row < n) {
        const float gi = acc_i[r] + Bi;
        const float gg = acc_g[r] + Bg;
        const float go = acc_o[r] + Bo;
        const float cv = sigmoidf_(gi) * tanhf(gg);
        h[(size_t)row * HDIM + jcol] = sigmoidf_(go) * tanhf(cv);
      }
    }
  }
}

// ---------------------------------------------------------------------------
// [N,128] x [128,128]^T WMMA GEMM with LDS-staged weights.
// Wt is [J=128, K=128] row-major (i.e. out = A @ Wt^T).
// The whole 64KB weight tile is staged once per block into LDS; B fragments
// are then ds_load_b64 reads.  Swizzle phys_k = (k + 4*j) & 127 keeps all 32
// lane bank-pairs of a b64 read disjoint (low lanes: banks 4j+ks, high lanes:
// 4j+ks+2) and is pair-preserving/even, so float2 stays 8B-aligned.
// RELU_A: relu on A load. BIAS/RELU_OUT applied at store.
// ---------------------------------------------------------------------------
template <bool RELU_A, bool BIAS, bool RELU_OUT>
__global__ __launch_bounds__(256) void gemm128_kernel(
    const float* __restrict__ A, const float* __restrict__ Wt,
    const float* __restrict__ bias, float* __restrict__ out, int n) {
  __shared__ float Bs[128 * 128];                 // exactly 64KB -> >=4 blocks/WGP

  // Cooperative stage: 256 threads x 16 float4 = 16384 floats.
  for (int idx = threadIdx.x; idx < 128 * 32; idx += 256) {
    const int j = idx >> 5;                       // weight row (output col) 0..127
    const int q = idx & 31;                       // float4 index within row
    const float4 v = *(const float4*)(Wt + (size_t)j * 128 + q * 4);
    const int pk = (q * 4 + 4 * j) & 127;         // swizzled dword offset
    *(float4*)(&Bs[j * 128 + pk]) = v;
  }
  __syncthreads();

  const int lane = threadIdx.x & 31;
  const int wave = threadIdx.x >> 5;
  const int rowbase = blockIdx.x * 128 + wave * 16;
  const int mrow  = lane & 15;
  const int khalf = (lane >> 4) << 1;
  int arow = rowbase + mrow;
  if (arow >= n) arow = n - 1;                    // clamp loads; stores guarded
  const float* aptr = A + (size_t)arow * HDIM + khalf;

  v8f acc[8];
  const v8f vzero = {};
  #pragma unroll
  for (int t = 0; t < 8; ++t) acc[t] = vzero;

  for (int k = 0; k < HDIM; k += 4) {
    v2f a = *(const v2f*)(aptr + k);
    if (RELU_A) { a.x = fmaxf(a.x, 0.0f); a.y = fmaxf(a.y, 0.0f); }
    const int ks = k + khalf;
    #pragma unroll
    for (int t = 0; t < 8; ++t) {
      const int j = t * 16 + mrow;
      const int pk = (ks + 4 * j) & 127;
      v2f b = *(const v2f*)(&Bs[j * 128 + pk]);
      acc[t] = wmma_f32_k4(a, b, acc[t]);
    }
  }

  const int mofs = (lane >> 4) << 3;
  #pragma unroll
  for (int t = 0; t < 8; ++t) {
    const int j = t * 16 + mrow;
    const float bb = BIAS ? bias[j] : 0.0f;
    #pragma unroll
    for (int r = 0; r < 8; ++r) {
      const int row = rowbase + r + mofs;
      if (row < n) {
        float v = acc[t][r] + bb;
        if (RELU_OUT) v = fmaxf(v, 0.0f);
        out[(size_t)row * HDIM + j] = v;
      }
    }
  }
}

// ---------------------------------------------------------------------------
// Helpers
// ---------------------------------------------------------------------------
// WT[j,k] = W[k,j] for 128x128 (so x @ W becomes the A @ WT^T fast path).
__global__ __launch_bounds__(256) void transpose128_kernel(
    const float* __restrict__ W, float* __restrict__ WT) {
  int gid = blockIdx.x * blockDim.x + threadIdx.x;
  if (gid < 128 * 128) {
    const int j = gid >> 7;
    const int k = gid & 127;
    WT[gid] = W[k * 128 + j];
  }
}

__global__ __launch_bounds__(256) void fill_ones_kernel(float* p, int n) {
  int i = blockIdx.x * blockDim.x + threadIdx.x;
  if (i < n) p[i] = 1.0f;                         // self-loop contributes 1
}

__global__ __launch_bounds__(256) void deg_kernel(const int* __restrict__ ei,
                                                  float* __restrict__ deg, int E) {
  int e = blockIdx.x * blockDim.x + threadIdx.x;
  if (e < E) atomicAdd(&deg[ei[E + e]], 1.0f);    // dst row of edge_index
}

__global__ __launch_bounds__(256) void rsqrt_kernel(float* p, int n) {
  int i = blockIdx.x * blockDim.x + threadIdx.x;
  if (i < n) p[i] = rsqrtf(p[i]);                 // deg >= 1 always
}

// x[n,j] = bias[j] + t[n,j] * dinv[n]^2   (self-loop term + bias init)
__global__ __launch_bounds__(256) void selfloop_bias_kernel(
    const float* __restrict__ t, const float* __restrict__ dinv,
    const float* __restrict__ bias, float* __restrict__ x, int total) {
  int gid = blockIdx.x * blockDim.x + threadIdx.x;
  if (gid < total) {
    const int node = gid >> 7;
    const int j = gid & 127;
    const float di = dinv[node];
    x[gid] = bias[j] + t[gid] * di * di;
  }
}

// One wave per edge; each lane owns 4 consecutive channels (float4).
// Feature rows live in the 192MB L2, so gathers + f32 atomics resolve on-chip.
__global__ __launch_bounds__(256) void scatter_kernel(
    const int* __restrict__ ei, const float* __restrict__ t,
    const float* __restrict__ dinv, float* __restrict__ x, int E) {
  int gid = blockIdx.x * blockDim.x + threadIdx.x;
  const int e = gid >> 5;
  if (e >= E) return;
  const int lane = gid & 31;
  const int s = ei[e];
  const int d = ei[E + e];
  const float nrm = dinv[s] * dinv[d];
  const float4 v = *(const float4*)(t + (size_t)s * HDIM + lane * 4);
  float* dst = x + (size_t)d * HDIM + lane * 4;
  atomicAdd(dst + 0, v.x * nrm);
  atomicAdd(dst + 1, v.y * nrm);
  atomicAdd(dst + 2, v.z * nrm);
  atomicAdd(dst + 3, v.w * nrm);
}

// ---------------------------------------------------------------------------
extern "C" void kernel_launch(void* const* d_in, const int* in_sizes, int n_in,
                              void* d_out, int out_size, void* d_ws, size_t ws_size,
                              hipStream_t stream) {
  const float* z    = (const float*)d_in[0];
  const int*   ei   = (const int*)d_in[1];
  const float* Wih  = (const float*)d_in[2];
  // d_in[3] = W_hh (unused: h0 = 0)
  const float* bih  = (const float*)d_in[4];
  const float* bhh  = (const float*)d_in[5];
  const float* W1   = (const float*)d_in[6];
  const float* b1   = (const float*)d_in[7];
  const float* W2   = (const float*)d_in[8];
  const float* b2   = (const float*)d_in[9];
  const float* W3   = (const float*)d_in[10];
  const float* b3   = (const float*)d_in[11];
  float* out = (float*)d_out;

  const int N = in_sizes[0] / HDIM;
  const int E = in_sizes[1] / 2;

  // Workspace: dinv[Npad] | buf0[N*128] | buf1[N*128] | W1T[16K] | W2T[16K]
  float* dinv = (float*)d_ws;
  const size_t npad = ((size_t)N + 3) & ~(size_t)3;   // keep float4 alignment
  float* buf0 = dinv + npad;
  float* buf1 = buf0 + (size_t)N * HDIM;
  float* w1t  = buf1 + (size_t)N * HDIM;
  float* w2t  = w1t + 128 * 128;

  const int total = N * HDIM;
  const dim3 blk(256);
  const dim3 grN((N + 255) / 256);
  const dim3 grE((E + 255) / 256);
  const dim3 grRows((N + 127) / 128);
  const dim3 grElem((total + 255) / 256);
  const dim3 grScat((E * 32 + 255) / 256);
  const dim3 grT(64);

  // weight transposes (tiny, once per launch)
  transpose128_kernel<<<grT, blk, 0, stream>>>(W1, w1t);
  transpose128_kernel<<<grT, blk, 0, stream>>>(W2, w2t);

  // degree -> dinv
  fill_ones_kernel<<<grN, blk, 0, stream>>>(dinv, N);
  deg_kernel<<<grE, blk, 0, stream>>>(ei, dinv, E);
  rsqrt_kernel<<<grN, blk, 0, stream>>>(dinv, N);

  // h = LSTM(z)                      -> buf0
  lstm_kernel<<<grRows, blk, 0, stream>>>(z, Wih, bih, bhh, buf0, N);

  // conv1: t = h @ W1                -> buf1
  gemm128_kernel<false, false, false>
      <<<grRows, blk, 0, stream>>>(buf0, w1t, nullptr, buf1, N);
  // x1 = b1 + t*dinv^2               -> buf0 ; then scatter edges t -> x1
  selfloop_bias_kernel<<<grElem, blk, 0, stream>>>(buf1, dinv, b1, buf0, total);
  scatter_kernel<<<grScat, blk, 0, stream>>>(ei, buf1, dinv, buf0, E);

  // conv2: t = relu(x1) @ W2         -> buf1   (ReLU fused into A load)
  gemm128_kernel<true, false, false>
      <<<grRows, blk, 0, stream>>>(buf0, w2t, nullptr, buf1, N);
  // x2 = b2 + t*dinv^2               -> buf0 ; scatter t -> x2
  selfloop_bias_kernel<<<grElem, blk, 0, stream>>>(buf1, dinv, b2, buf0, total);
  scatter_kernel<<<grScat, blk, 0, stream>>>(ei, buf1, dinv, buf0, E);

  // out = relu(x2 @ W3^T + b3)   (W3 already stored [J,K])
  gemm128_kernel<false, true, true>
      <<<grRows, blk, 0, stream>>>(buf0, W3, b3, out, N);
}